// GRAPEOptimizer_66391604461568
// MI455X (gfx1250) — compile-verified
//
#include <hip/hip_runtime.h>

typedef __attribute__((ext_vector_type(2))) float v2f;
typedef __attribute__((ext_vector_type(8))) float v8f;

#if __has_builtin(__builtin_amdgcn_wmma_f32_16x16x4_f32)
#define HAVE_WMMA_F32_4 1
#else
#define HAVE_WMMA_F32_4 0
#endif

// Kernel A: compose the nsteps commuting X-rotations into one rotation.
// Theta = (sum a_k) * dt/2 ; c = cos Theta, s = sin Theta (double precision, once).
__global__ void grape_compose(const float* __restrict__ amp, int nsteps,
                              double dt_half, float* __restrict__ cs) {
    if (blockIdx.x == 0 && threadIdx.x == 0) {
        double th = 0.0;
        for (int k = 0; k < nsteps; ++k) th += (double)amp[k];
        th *= dt_half;
        cs[0] = (float)cos(th);
        cs[1] = (float)sin(th);
    }
}

// Kernel B: apply the composed 4x4 real rotation to every column via
// V_WMMA_F32_16X16X4_F32 (16 columns/WMMA, 4 tiles unrolled per wave iteration),
// streaming with non-temporal hints at HBM rate.
//
// Per-column update (v = [s0r, s0i, s1r, s1i]):
//   s0r' =  c*s0r + s*s1i
//   s0i' =  c*s0i - s*s1r
//   s1r' =  s*s0i + c*s1r
//   s1i' = -s*s0r + c*s1i
__global__ void __launch_bounds__(256)
grape_apply(const float* __restrict__ sr, const float* __restrict__ si,
            const float* __restrict__ cs, float* __restrict__ out,
            long long B, int numTiles) {
    const float c = cs[0];
    const float s = cs[1];

    const int gid    = blockIdx.x * blockDim.x + threadIdx.x;
    const int lane   = gid & 31;
    const int wave   = gid >> 5;
    const int nWaves = (gridDim.x * blockDim.x) >> 5;

    const int  m  = lane & 15;
    const bool hi = lane >= 16;

#if HAVE_WMMA_F32_4
    // A-matrix (16x4 f32, wave32): lane m -> {A[m][0], A[m][1]},
    // lane 16+m -> {A[m][2], A[m][3]}. Rows 4..15 are zero.
    // M = [[ c, 0, 0, s],
    //      [ 0, c,-s, 0],
    //      [ 0, s, c, 0],
    //      [-s, 0, 0, c]]
    float a0 = 0.0f, a1 = 0.0f;
    switch (lane) {
        case 0:  a0 =  c; break;   // A[0][0]
        case 1:  a1 =  c; break;   // A[1][1]
        case 2:  a1 =  s; break;   // A[2][1]
        case 3:  a0 = -s; break;   // A[3][0]
        case 16: a1 =  s; break;   // A[0][3]
        case 17: a0 = -s; break;   // A[1][2]
        case 18: a0 =  c; break;   // A[2][2]
        case 19: a1 =  c; break;   // A[3][3]
        default: break;
    }
    v2f Amat; Amat.x = a0; Amat.y = a1;

    // Per-lane base pointers (hoisted out of the loop):
    // B-matrix (4x16 f32): lanes 0-15 hold K=0,1 (row-0 re/im),
    // lanes 16-31 hold K=2,3 (row-1 re/im).
    const float* __restrict__ pr = (hi ? (sr + B) : sr) + m;
    const float* __restrict__ pi = (hi ? (si + B) : si) + m;
    float* __restrict__ o0 = out + m;              // re row0
    float* __restrict__ o1 = out + B + m;          // re row1
    float* __restrict__ o2 = out + 2 * B + m;      // im row0
    float* __restrict__ o3 = out + 3 * B + m;      // im row1

    const int numGroups = numTiles >> 2;           // 4 tiles (64 cols) per group
    for (int g = wave; g < numGroups; g += nWaves) {
        const long long base = (long long)g << 6;  // g * 64 columns

        // Issue all 8 NT loads back-to-back for deep MLP.
        v2f Bm0, Bm1, Bm2, Bm3;
        Bm0.x = __builtin_nontemporal_load(pr + base);
        Bm1.x = __builtin_nontemporal_load(pr + base + 16);
        Bm2.x = __builtin_nontemporal_load(pr + base + 32);
        Bm3.x = __builtin_nontemporal_load(pr + base + 48);
        Bm0.y = __builtin_nontemporal_load(pi + base);
        Bm1.y = __builtin_nontemporal_load(pi + base + 16);
        Bm2.y = __builtin_nontemporal_load(pi + base + 32);
        Bm3.y = __builtin_nontemporal_load(pi + base + 48);

        v8f C0 = {}, C1 = {}, C2 = {}, C3 = {};
        C0 = __builtin_amdgcn_wmma_f32_16x16x4_f32(false, Amat, false, Bm0, (short)0, C0, false, false);
        C1 = __builtin_amdgcn_wmma_f32_16x16x4_f32(false, Amat, false, Bm1, (short)0, C1, false, false);
        C2 = __builtin_amdgcn_wmma_f32_16x16x4_f32(false, Amat, false, Bm2, (short)0, C2, false, false);
        C3 = __builtin_amdgcn_wmma_f32_16x16x4_f32(false, Amat, false, Bm3, (short)0, C3, false, false);

        if (!hi) {
            // D VGPR j, lanes 0-15 = row M=j.
            __builtin_nontemporal_store(C0[0], o0 + base);
            __builtin_nontemporal_store(C0[2], o1 + base);
            __builtin_nontemporal_store(C0[1], o2 + base);
            __builtin_nontemporal_store(C0[3], o3 + base);
            __builtin_nontemporal_store(C1[0], o0 + base + 16);
            __builtin_nontemporal_store(C1[2], o1 + base + 16);
            __builtin_nontemporal_store(C1[1], o2 + base + 16);
            __builtin_nontemporal_store(C1[3], o3 + base + 16);
            __builtin_nontemporal_store(C2[0], o0 + base + 32);
            __builtin_nontemporal_store(C2[2], o1 + base + 32);
            __builtin_nontemporal_store(C2[1], o2 + base + 32);
            __builtin_nontemporal_store(C2[3], o3 + base + 32);
            __builtin_nontemporal_store(C3[0], o0 + base + 48);
            __builtin_nontemporal_store(C3[2], o1 + base + 48);
            __builtin_nontemporal_store(C3[1], o2 + base + 48);
            __builtin_nontemporal_store(C3[3], o3 + base + 48);
        }
    }

    // Leftover whole tiles (numTiles % 4), one tile per wave iteration.
    for (int t = (numGroups << 2) + wave; t < numTiles; t += nWaves) {
        const long long col = (long long)t << 4;   // t * 16 columns
        v2f Bm;
        Bm.x = __builtin_nontemporal_load(pr + col);
        Bm.y = __builtin_nontemporal_load(pi + col);
        v8f C = {};
        C = __builtin_amdgcn_wmma_f32_16x16x4_f32(false, Amat, false, Bm, (short)0, C, false, false);
        if (!hi) {
            __builtin_nontemporal_store(C[0], o0 + col);
            __builtin_nontemporal_store(C[2], o1 + col);
            __builtin_nontemporal_store(C[1], o2 + col);
            __builtin_nontemporal_store(C[3], o3 + col);
        }
    }
    const long long tailBase = (long long)numTiles * 16;
#else
    const long long tailBase = 0;  // no WMMA builtin: do everything elementwise
#endif

    // Tail (B not a multiple of 16) — plain VALU path.
    const long long stride = (long long)gridDim.x * blockDim.x;
    for (long long idx = tailBase + gid; idx < B; idx += stride) {
        const float s0r = sr[idx], s1r = sr[B + idx];
        const float s0i = si[idx], s1i = si[B + idx];
        out[idx]         = c * s0r + s * s1i;
        out[B + idx]     = c * s1r + s * s0i;
        out[2 * B + idx] = c * s0i - s * s1r;
        out[3 * B + idx] = c * s1i - s * s0r;
    }
}

extern "C" void kernel_launch(void* const* d_in, const int* in_sizes, int n_in,
                              void* d_out, int out_size, void* d_ws, size_t ws_size,
                              hipStream_t stream) {
    const float* amp = (const float*)d_in[0];
    const float* sr  = (const float*)d_in[1];
    const float* si  = (const float*)d_in[2];
    float* out = (float*)d_out;
    float* cs  = (float*)d_ws;  // 2 floats: {cos, sin}

    const int nsteps = in_sizes[0];
    const long long B = (long long)in_sizes[1] / 2;
    const double dt_half = 0.5 / (double)(nsteps > 0 ? nsteps : 1);

    grape_compose<<<1, 32, 0, stream>>>(amp, nsteps, dt_half, cs);

    const int numTiles = (int)(B / 16);
    const int threads  = 256;                    // 8 wave32 per block
    int blocks = 2048;                           // 16384 waves -> ~8 groups/wave at B=8.4M
    const int groupsPerWaveTarget = 8;
    long long groups = numTiles >> 2;
    long long wavesWanted = (groups + groupsPerWaveTarget - 1) / groupsPerWaveTarget;
    long long blocksWanted = (wavesWanted + (threads / 32) - 1) / (threads / 32);
    if (blocksWanted < 1) blocksWanted = 1;
    if (blocksWanted < blocks) blocks = (int)blocksWanted;

    grape_apply<<<blocks, threads, 0, stream>>>(sr, si, cs, out, B, numTiles);
}